// TemporalISWT_78400333021538
// MI455X (gfx1250) — compile-verified
//
#include <hip/hip_runtime.h>

// Inverse stationary Haar wavelet, 3 levels, fused into one 8-tap composite
// FIR + transpose [B,N,C,T] -> [B,T,N,C].
//
// MI455X reasoning: 640 MB mandatory traffic @ 23.3 TB/s => ~27us floor;
// 0.7 GFLOP of VALU work is negligible. Memory-bound => use CDNA5 async
// global->LDS copies (ASYNCcnt) to stage tiles, LDS for the transpose so
// loads (contiguous in t) AND stores (contiguous in c) are both coalesced,
// and non-temporal stores for the write-once output.

#define B_   8
#define N_   128
#define C_   32
#define T_   1024
#define TT   128                 // t-tile per block
#define HALO 8                   // composite filter support (8 taps); look-back is 7
#define COLS (TT + HALO)         // 136 valid columns per LDS row
#define RS   140                 // padded row stride (floats): 16B aligned
#define NVEC (COLS / 4)          // 34 b128 transfers per row

__device__ __forceinline__ void async_ld16(unsigned ldsByteAddr, const void* gaddr) {
  // GLOBAL_LOAD_ASYNC_TO_LDS_B128: per-lane 16B copy from global to LDS, tracked by ASYNCcnt.
  asm volatile("global_load_async_to_lds_b128 %0, %1, off"
               :: "v"(ldsByteAddr), "v"(gaddr)
               : "memory");
}

__global__ __launch_bounds__(256) void iswt_haar3_kernel(
    const float* __restrict__ ca, const float* __restrict__ cd3,
    const float* __restrict__ cd2, const float* __restrict__ cd1,
    float* __restrict__ out) {
  extern __shared__ float sm[];  // 4 inputs * 32 rows * RS floats = 71,680 B

  const int tid  = threadIdx.x;
  const int tile = blockIdx.x;   // t-tile
  const int n    = blockIdx.y;
  const int b    = blockIdx.z;
  const int t0   = tile * TT;
  const int ln   = b * N_ + n;
  const unsigned smBase = (unsigned)(size_t)(void*)sm;  // flat low 32b = LDS offset

  // ---- Stage all 4 inputs' (32 x 136) tile into LDS via async b128 copies.
  const float* srcs[4] = {ca, cd3, cd2, cd1};
#pragma unroll
  for (int i = 0; i < 4; ++i) {
    const float* src = srcs[i];
    for (int v = tid; v < C_ * NVEC; v += 256) {
      int c = v / NVEC;
      int j = v - c * NVEC;            // which b128 within the row
      int tg = t0 - HALO + 4 * j;      // global t of first element
      if (tg < 0) tg = 0;              // clamp; halo re-zeroed below for t0==0
      const float* gp = src + ((size_t)(ln * C_ + c) * T_ + tg);
      unsigned lo = smBase + (unsigned)(((i * C_ + c) * RS + 4 * j) * 4);
      async_ld16(lo, gp);
    }
  }
  asm volatile("s_wait_asynccnt 0" ::: "memory");
  __syncthreads();                      // every wave waited its own ASYNCcnt

  // Zero-padding semantics at the sequence start: zero the 8 halo columns.
  if (t0 == 0) {
    for (int z = tid; z < 4 * C_ * HALO; z += 256) {
      int i = z >> 8;                   // input
      int r = z & 255;
      int c = r >> 3;
      int k = r & 7;                    // halo column 0..7
      sm[(i * C_ + c) * RS + k] = 0.0f;
    }
  }
  __syncthreads();

  // Composite Haar synthesis coefficients (s = 1/sqrt(2)).
  const float K3 = 0.04419417382415922f;   // s^3 / 8  (ca, cd3 taps)
  const float K2 = 0.125f;                 // s^2 / 4  (cd2 taps)
  const float K1 = 0.35355339059327373f;   // s   / 2  (cd1 taps)

  // ---- Compute + transposed store. lane = channel c (wave32 == C), each
  // group of 32 lanes handles 16 consecutive t; store is 128B coalesced.
  const int c = tid & 31;
  const int g = tid >> 5;                  // 0..7
  const float* A  = sm + (0 * C_ + c) * RS;
  const float* D3 = sm + (1 * C_ + c) * RS;
  const float* D2 = sm + (2 * C_ + c) * RS;
  const float* D1 = sm + (3 * C_ + c) * RS;
  float* op = out + (((size_t)b * T_ + (t0 + g * 16)) * N_ + n) * C_ + c;

#pragma unroll
  for (int uu = 0; uu < 16; ++uu) {
    const int tl = g * 16 + uu + HALO;     // LDS column for output t = t0+g*16+uu
    float sca = 0.0f, sd3 = 0.0f;
#pragma unroll
    for (int k = 0; k < 8; ++k) sca += A[tl - k];
#pragma unroll
    for (int k = 0; k < 4; ++k) sd3 += D3[tl - k] - D3[tl - 4 - k];
    float val = K3 * (sca + sd3)
              + K2 * (D2[tl] + D2[tl - 1] - D2[tl - 2] - D2[tl - 3])
              + K1 * (D1[tl] - D1[tl - 1]);
    __builtin_nontemporal_store(val, op);
    op += (size_t)N_ * C_;                 // next t, same (n, c)
  }
}

extern "C" void kernel_launch(void* const* d_in, const int* in_sizes, int n_in,
                              void* d_out, int out_size, void* d_ws, size_t ws_size,
                              hipStream_t stream) {
  (void)in_sizes; (void)n_in; (void)out_size; (void)d_ws; (void)ws_size;
  const float* ca  = (const float*)d_in[0];
  const float* cd3 = (const float*)d_in[1];
  const float* cd2 = (const float*)d_in[2];
  const float* cd1 = (const float*)d_in[3];
  float* out = (float*)d_out;

  const size_t shmem = (size_t)4 * C_ * RS * sizeof(float);  // 71,680 B
  hipFuncSetAttribute(reinterpret_cast<const void*>(iswt_haar3_kernel),
                      hipFuncAttributeMaxDynamicSharedMemorySize, (int)shmem);

  dim3 grid(T_ / TT, N_, B_);   // (8, 128, 8) = 8192 blocks
  dim3 block(256);              // 8 waves of 32
  iswt_haar3_kernel<<<grid, block, shmem, stream>>>(ca, cd3, cd2, cd1, out);
}